// FasterRCNN_9835475108289
// MI455X (gfx1250) — compile-verified
//
#include <hip/hip_runtime.h>
#include <hip/hip_bf16.h>

#define GT_CAP   128
#define NMS_CAP  2048
#define IOU_LOW  0.3f
#define IOU_HIGH 0.6f
#define NEUTRAL_VAL -1e8f

typedef float v2f __attribute__((ext_vector_type(2)));
typedef float v8f __attribute__((ext_vector_type(8)));

__device__ __forceinline__ float fast_rcp(float x) {
#if __has_builtin(__builtin_amdgcn_rcpf)
    return __builtin_amdgcn_rcpf(x);   // v_rcp_f32 (1 TRANS op)
#else
    return 1.0f / x;
#endif
}

__device__ __forceinline__ void wave_fence() {
#if __has_builtin(__builtin_amdgcn_wave_barrier)
    __builtin_amdgcn_wave_barrier();
#endif
}

// ---------------------------------------------------------------------------
// Kernel 1: anchor->GT matching.
// One wave32 = 16 anchors x 16 GT tile. IoU denominator outer-sum
// area_a[i]+area_g[j] via V_WMMA_F32_16X16X4_F32 (A=[areaA,1], B=[1;areaG]);
// intersection min/max per-lane on the WMMA C/D layout (lanes 0-15: M=0..7,
// lanes 16-31: M=8..15, N=lane&15). GT staged via gfx1250 async-to-LDS DMA.
// CT_TILES: compile-time GT tile count (0 = runtime loop).
// ---------------------------------------------------------------------------
template<int CT_TILES>
__global__ __launch_bounds__(256) void anchor_match_kernel(
    const float* __restrict__ anchors, const float* __restrict__ gt,
    float* __restrict__ out, int nA, int nG)
{
    __shared__ float sgt[GT_CAP * 5];     // GT rows (x1,y1,x2,y2,cls)
    __shared__ float sga[GT_CAP];         // GT areas
    __shared__ float sanch[8][16][4];     // per-wave anchor coords
    __shared__ float smout[8][80];        // per-wave staged output (16 rows x 5)

    const int t    = threadIdx.x;
    const int lane = t & 31;
    const int wid  = t >> 5;
    const int gtTiles = CT_TILES ? CT_TILES : ((nG + 15) >> 4);
    const int gtot    = nG * 5;

    // zero-pad tail (disjoint from async-written region -> no ordering hazard)
    for (int i = gtot + t; i < GT_CAP * 5; i += 256) sgt[i] = 0.0f;

#if defined(__gfx1250__)
    // CDNA5 async DMA: global -> LDS, tracked with ASYNCcnt
    for (int i = t; i < gtot; i += 256) {
        unsigned lds_off = (unsigned)(unsigned long long)(&sgt[i]);
        const float* gp = gt + i;
        asm volatile("global_load_async_to_lds_b32 %0, %1, off"
                     :: "v"(lds_off), "v"(gp) : "memory");
    }
    asm volatile("s_wait_asynccnt 0" ::: "memory");
#else
    for (int i = t; i < gtot; i += 256) sgt[i] = gt[i];
#endif
    __syncthreads();

    if (t < GT_CAP) {
        float x1 = sgt[t*5+0], y1 = sgt[t*5+1], x2 = sgt[t*5+2], y2 = sgt[t*5+3];
        sga[t] = (x2 - x1) * (y2 - y1);
    }

    const int anchorBase = (blockIdx.x * 8 + wid) * 16;
    float myArea = 0.0f;
    if (anchorBase < nA) {
        int a = anchorBase + (lane & 15);
        if (a >= nA) a = nA - 1;                      // clamp (tiles wave-uniform)
        float ax1 = anchors[a*4+0], ay1 = anchors[a*4+1];
        float ax2 = anchors[a*4+2], ay2 = anchors[a*4+3];
        myArea = (ax2 - ax1) * (ay2 - ay1);
        if (lane < 16) {
            sanch[wid][lane][0] = ax1; sanch[wid][lane][1] = ay1;
            sanch[wid][lane][2] = ax2; sanch[wid][lane][3] = ay2;
        }
    }
    __syncthreads();

    if (anchorBase >= nA) return;   // wave-uniform; EXEC stays all-1s below

    const int mbase = (lane < 16) ? 0 : 8;
    const int ln    = lane & 15;

    // hoist tile-invariant anchor coords: 32 LDS loads total
    float rx1[8], ry1[8], rx2[8], ry2[8];
#pragma unroll
    for (int r = 0; r < 8; ++r) {
        rx1[r] = sanch[wid][mbase + r][0];
        ry1[r] = sanch[wid][mbase + r][1];
        rx2[r] = sanch[wid][mbase + r][2];
        ry2[r] = sanch[wid][mbase + r][3];
    }

    float best[8]; int bestj[8];
#pragma unroll
    for (int r = 0; r < 8; ++r) { best[r] = -1.0f; bestj[r] = 0; }

#if __has_builtin(__builtin_amdgcn_wmma_f32_16x16x4_f32)
    // A (16x4 f32): lanes 0-15 hold {K0=areaA[M], K1=1}; lanes 16-31: K2,K3=0
    v2f afrag;
    afrag.x = (lane < 16) ? myArea : 0.0f;
    afrag.y = (lane < 16) ? 1.0f   : 0.0f;
#endif

    // gArea loaded UNCONDITIONALLY by all lanes (no exec-masked LDS load);
    // the lane<16 select below becomes a branchless v_cndmask.
    auto tile_body = [&](int tb, float gArea) {
        const int jn = (tb << 4) + ln;     // GT column owned by this lane
        v8f dmat;
#if __has_builtin(__builtin_amdgcn_wmma_f32_16x16x4_f32)
        // B (4x16): row K0 = ones, K1 = areaG  ->  D[i][j] = areaA[i]+areaG[j]
        v2f bfrag;
        bfrag.x = (lane < 16) ? 1.0f : gArea;
        bfrag.y = 0.0f;
        v8f cz = {0.f,0.f,0.f,0.f,0.f,0.f,0.f,0.f};
        dmat = __builtin_amdgcn_wmma_f32_16x16x4_f32(
                   false, afrag, false, bfrag, (short)0, cz, false, false);
#else
        (void)gArea;
        float myGA = sga[jn];
#pragma unroll
        for (int r = 0; r < 8; ++r)
            dmat[r] = (rx2[r]-rx1[r]) * (ry2[r]-ry1[r]) + myGA;
#endif
        const float gx1 = sgt[jn*5+0], gy1 = sgt[jn*5+1];
        const float gx2 = sgt[jn*5+2], gy2 = sgt[jn*5+3];
#pragma unroll
        for (int r = 0; r < 8; ++r) {
            float w = fminf(rx2[r], gx2) - fmaxf(rx1[r], gx1); w = fmaxf(w, 0.0f);
            float h = fminf(ry2[r], gy2) - fmaxf(ry1[r], gy1); h = fmaxf(h, 0.0f);
            float inter = w * h;
            float iou = inter * fast_rcp(dmat[r] - inter);   // inter/(a1+a2-inter)
            if (iou > best[r]) { best[r] = iou; bestj[r] = jn; }
        }
    };

    if (CT_TILES) {
        // prefetch B-operand areas for all tiles in one LDS clause
        float ga[CT_TILES ? CT_TILES : 1];
#pragma unroll
        for (int tb = 0; tb < CT_TILES; ++tb) ga[tb] = sga[(tb << 4) + ln];
#pragma unroll
        for (int tb = 0; tb < CT_TILES; ++tb) tile_body(tb, ga[tb]);
    } else {
        for (int tb = 0; tb < gtTiles; ++tb) tile_body(tb, sga[(tb << 4) + ln]);
    }

    // row max/argmax: reduce across the 16 lanes of each half (masks < 16)
#pragma unroll
    for (int r = 0; r < 8; ++r) {
        float b = best[r]; int bj = bestj[r];
#pragma unroll
        for (int m = 1; m < 16; m <<= 1) {
            float ob  = __shfl_xor(b,  m, 32);
            int   obj = __shfl_xor(bj, m, 32);
            if (ob > b || (ob == b && obj < bj)) { b = ob; bj = obj; }
        }
        best[r] = b; bestj[r] = bj;
    }

    // stage the wave's 16 rows x 5 outputs in LDS, then store coalesced
    if (ln == 0) {  // lanes 0 (rows 0-7) and 16 (rows 8-15)
#pragma unroll
        for (int r = 0; r < 8; ++r) {
            float q = best[r]; int bj = bestj[r];
            float o[5];
            if (q <= IOU_LOW)       { for (int c = 0; c < 5; ++c) o[c] = -1.0f; }
            else if (q < IOU_HIGH)  { for (int c = 0; c < 5; ++c) o[c] = NEUTRAL_VAL; }
            else                    { for (int c = 0; c < 5; ++c) o[c] = sgt[bj*5+c]; }
            for (int c = 0; c < 5; ++c) smout[wid][(mbase + r) * 5 + c] = o[c];
        }
    }
    wave_fence();   // wave-local LDS ordering (ds ops complete in order)

    float* dst = out + (size_t)anchorBase * 5;
    if (anchorBase + 16 <= nA) {
        // 80 contiguous floats per wave -> 20 lanes x global_store_b128
        if (lane < 20) {
            float4 v = *(const float4*)&smout[wid][lane * 4];
            *(float4*)(dst + lane * 4) = v;
        }
    } else {
        int lim = (nA - anchorBase) * 5;
        for (int i = lane; i < lim; i += 32) dst[i] = smout[wid][i];
    }
}

// ---------------------------------------------------------------------------
// Kernel 2: class-spec NMS. Single workgroup (32 wave32s on one WGP); max
// reduction, bitonic sort (descending) and sequential suppression in LDS.
// Suppression test is division-free: iou >= 0.5  <=>  3*inter >= ai + aj.
// ---------------------------------------------------------------------------
__global__ __launch_bounds__(1024) void nms_kernel(
    const float* __restrict__ pb, const float* __restrict__ ps,
    const int* __restrict__ cid, float* __restrict__ out, int P)
{
    __shared__ float skey[NMS_CAP];
    __shared__ int   sidx[NMS_CAP];
    __shared__ float bx1[NMS_CAP], by1[NMS_CAP], bx2[NMS_CAP], by2[NMS_CAP], bar[NMS_CAP];
    __shared__ unsigned char ssup[NMS_CAP];

    const int t = threadIdx.x;

    // ---- max over all prop_boxes elements -> per-class shift unit ----
    float lm = -3.402823466e38f;
    for (int i = t; i < P * 4; i += 1024) lm = fmaxf(lm, pb[i]);
    skey[t] = lm;
    __syncthreads();
    for (int s = 512; s > 0; s >>= 1) {
        if (t < s) skey[t] = fmaxf(skey[t], skey[t + s]);
        __syncthreads();
    }
    const float shiftUnit = skey[0] + 1.0f;
    __syncthreads();

    // ---- bitonic sort (score descending, payload = original index) ----
    for (int i = t; i < NMS_CAP; i += 1024) {
        if (i < P) { skey[i] = ps[i]; sidx[i] = i; }
        else       { skey[i] = -3.402823466e38f; sidx[i] = i; }
    }
    for (int k = 2; k <= NMS_CAP; k <<= 1)
        for (int j = k >> 1; j > 0; j >>= 1) {
            __syncthreads();
            int i = ((t & ~(j - 1)) << 1) | (t & (j - 1));
            int p = i | j;
            bool desc = ((i & k) == 0);
            float a = skey[i], b = skey[p];
            bool sw = desc ? (a < b) : (a > b);
            if (sw) {
                skey[i] = b; skey[p] = a;
                int q = sidx[i]; sidx[i] = sidx[p]; sidx[p] = q;
            }
        }
    __syncthreads();

    // ---- gather class-offset boxes in sorted order (SoA) ----
    for (int i = t; i < NMS_CAP; i += 1024) {
        if (i < P) {
            int o = sidx[i];
            float off = (float)cid[o] * shiftUnit;
            float x1 = pb[o*4+0] + off, y1 = pb[o*4+1] + off;
            float x2 = pb[o*4+2] + off, y2 = pb[o*4+3] + off;
            bx1[i] = x1; by1[i] = y1; bx2[i] = x2; by2[i] = y2;
            bar[i] = (x2 - x1) * (y2 - y1);
        }
        ssup[i] = 0;
    }

    // ---- sequential greedy suppression (division-free test) ----
    for (int i = 0; i < P; ++i) {
        __syncthreads();
        if (ssup[i]) continue;                      // uniform read after barrier
        float xi1 = bx1[i], yi1 = by1[i], xi2 = bx2[i], yi2 = by2[i], ari = bar[i];
        for (int j = i + 1 + t; j < P; j += 1024) {
            float w = fminf(bx2[j], xi2) - fmaxf(bx1[j], xi1); w = fmaxf(w, 0.f);
            float h = fminf(by2[j], yi2) - fmaxf(by1[j], yi1); h = fmaxf(h, 0.f);
            float inter = w * h;
            // inter/(ai+aj-inter) >= 0.5  <=>  3*inter >= ai+aj   (denom > 0)
            if (3.0f * inter >= bar[j] + ari) ssup[j] = 1;
        }
    }
    __syncthreads();

    // ---- scatter kept scores back to original order ----
    for (int i = t; i < P; i += 1024) {
        int o = sidx[i];
        out[o] = ssup[i] ? 0.0f : skey[i];
    }
}

// ---------------------------------------------------------------------------
extern "C" void kernel_launch(void* const* d_in, const int* in_sizes, int n_in,
                              void* d_out, int out_size, void* d_ws, size_t ws_size,
                              hipStream_t stream) {
    (void)n_in; (void)out_size; (void)d_ws; (void)ws_size;
    const float* anchors = (const float*)d_in[0];
    const float* gt      = (const float*)d_in[1];
    const float* pb      = (const float*)d_in[2];
    const float* ps      = (const float*)d_in[3];
    const int*   cid     = (const int*)  d_in[4];
    float* out = (float*)d_out;

    const int nA = in_sizes[0] / 4;   // 250000
    const int nG = in_sizes[1] / 5;   // 128
    const int P  = in_sizes[3];       // 2000

    const int waves  = (nA + 15) / 16;
    const int blocks = (waves + 7) / 8;
    if (nG == 128) {  // specialized: 8 fully-unrolled GT tiles -> 8 static WMMAs
        anchor_match_kernel<8><<<blocks, 256, 0, stream>>>(anchors, gt, out, nA, nG);
    } else {
        anchor_match_kernel<0><<<blocks, 256, 0, stream>>>(anchors, gt, out, nA, nG);
    }
    nms_kernel<<<1, 1024, 0, stream>>>(pb, ps, cid, out + (size_t)nA * 5, P);
}